// LiftSplatShoot_3169685865331
// MI455X (gfx1250) — compile-verified
//
#include <hip/hip_runtime.h>

typedef __attribute__((ext_vector_type(2))) float v2f;
typedef __attribute__((ext_vector_type(8))) float v8f;

#define BSZ   4
#define NCAM  6
#define DD    41
#define FHh   16
#define FWw   44
#define CAMC  64
#define PTS_PER_BN (DD*FHh*FWw)        /* 28864 */
#define NPRIME (BSZ*NCAM*PTS_PER_BN)   /* 692736 */
#define NXV 200
#define NYV 200
#define CH_STRIDE (NXV*NYV)            /* 40000 */
#define OUT_PER_B (CAMC*CH_STRIDE)     /* 2560000 */

// ---------------- setup: per-(b,n) transform params -> ws -------------------
// layout per bn (24 floats):
//   [0..8]  inv(post_rots) row-major
//   [9..11] post_trans
//   [12..23] Caug = [rots @ inv(intrins) | trans]  row-major 3x4
__global__ void lss_setup(const float* __restrict__ rots,
                          const float* __restrict__ trans,
                          const float* __restrict__ intrins,
                          const float* __restrict__ post_rots,
                          const float* __restrict__ post_trans,
                          float* __restrict__ P) {
  int bn = threadIdx.x;
  if (bn >= BSZ * NCAM) return;
  const float* pr = post_rots + bn * 9;
  const float* K  = intrins   + bn * 9;
  const float* R  = rots      + bn * 9;
  float* o = P + bn * 24;
  {
    float a = pr[0], b = pr[1], c = pr[2], d = pr[3], e = pr[4], f = pr[5],
          g = pr[6], h = pr[7], i = pr[8];
    float A = e * i - f * h, Bc = -(d * i - f * g), Cc = d * h - e * g;
    float id = 1.0f / (a * A + b * Bc + c * Cc);
    o[0] = A * id;  o[1] = -(b * i - c * h) * id; o[2] =  (b * f - c * e) * id;
    o[3] = Bc * id; o[4] =  (a * i - c * g) * id; o[5] = -(a * f - c * d) * id;
    o[6] = Cc * id; o[7] = -(a * h - b * g) * id; o[8] =  (a * e - b * d) * id;
  }
  o[9]  = post_trans[bn * 3 + 0];
  o[10] = post_trans[bn * 3 + 1];
  o[11] = post_trans[bn * 3 + 2];
  float ki[9];
  {
    float a = K[0], b = K[1], c = K[2], d = K[3], e = K[4], f = K[5],
          g = K[6], h = K[7], i = K[8];
    float A = e * i - f * h, Bc = -(d * i - f * g), Cc = d * h - e * g;
    float id = 1.0f / (a * A + b * Bc + c * Cc);
    ki[0] = A * id;  ki[1] = -(b * i - c * h) * id; ki[2] =  (b * f - c * e) * id;
    ki[3] = Bc * id; ki[4] =  (a * i - c * g) * id; ki[5] = -(a * f - c * d) * id;
    ki[6] = Cc * id; ki[7] = -(a * h - b * g) * id; ki[8] =  (a * e - b * d) * id;
  }
  for (int m = 0; m < 3; ++m) {
    for (int n = 0; n < 3; ++n) {
      float s = 0.f;
      for (int k = 0; k < 3; ++k) s += R[m * 3 + k] * ki[k * 3 + n];
      o[12 + m * 4 + n] = s;
    }
    o[12 + m * 4 + 3] = trans[bn * 3 + m];
  }
}

// ---------------- zero output ----------------------------------------------
__global__ void lss_zero(float* __restrict__ out, int n4) {
  int i = blockIdx.x * blockDim.x + threadIdx.x;
  if (i < n4) ((float4*)out)[i] = make_float4(0.f, 0.f, 0.f, 0.f);
}

// ---------------- main: WMMA geometry + coalesced atomic scatter ------------
__global__ void __launch_bounds__(256)
lss_scatter(const float* __restrict__ x_img,
            const float* __restrict__ P,
            float* __restrict__ out) {
  const int lane = threadIdx.x & 31;
  const int wave = blockIdx.x * (blockDim.x >> 5) + (threadIdx.x >> 5);
  const int GROUPS_PER_BN = PTS_PER_BN / 16;  // 1804
  const int bn  = wave / GROUPS_PER_BN;
  const int grp = wave - bn * GROUPS_PER_BN;
  const float* p = P + bn * 24;

  // point slot m = lane&15; lanes 16..31 recompute the same point so that the
  // B-operand half-wave split (K=0,1 | K=2,3) needs no cross-lane movement.
  const int m = lane & 15;
  const int l = grp * 16 + m;
  const int d  = l / (FHh * FWw);
  const int rm = l - d * (FHh * FWw);
  const int h  = rm / FWw;
  const int w  = rm - h * FWw;
  const float u   = (float)w * (703.0f / 43.0f);
  const float vv  = (float)h * 17.0f;
  const float dep = 4.0f + (float)d;

  const float px = u   - p[9];
  const float py = vv  - p[10];
  const float pz = dep - p[11];
  const float qx = p[0] * px + p[1] * py + p[2] * pz;
  const float qy = p[3] * px + p[4] * py + p[5] * pz;
  const float qz = p[6] * px + p[7] * py + p[8] * pz;

  // B (4x16, KxN): vgpr0 = K0 | K2 , vgpr1 = K1 | K3 across half-waves
  v2f Bm;
  Bm.x = (lane < 16) ? qx * qz : qz;
  Bm.y = (lane < 16) ? qy * qz : 1.0f;

  // A (16x4): rows 0..2 = Caug, rows 3..15 = 0. lane = M row; half-wave = K pair
  const int col4 = (lane < 16) ? 0 : 2;
  const int mr   = (m < 3) ? m : 0;
  const float a0c = p[12 + mr * 4 + col4];
  const float a1c = p[13 + mr * 4 + col4];
  v2f Am;
  Am.x = (m < 3) ? a0c : 0.f;
  Am.y = (m < 3) ? a1c : 0.f;

  v8f acc = {};
  acc = __builtin_amdgcn_wmma_f32_16x16x4_f32(false, Am, false, Bm,
                                              (short)0, acc, false, false);
  // D[m,n]: rows 0..2 live in vgprs 0..2 of lanes n<16 -> each lane n holds
  // its own point's geom (x,y,z) in acc[0..2].
  const float gx = acc[0], gy = acc[1], gz = acc[2];
  const int ix = (int)((gx + 50.0f) * 2.0f);
  const int iy = (int)((gy + 50.0f) * 2.0f);
  const int iz = (int)((gz + 10.0f) * 0.05f);
  const int keep = (lane < 16) & (ix >= 0) & (ix < NXV) &
                   (iy >= 0) & (iy < NYV) & (iz == 0);
  const int b    = bn / NCAM;
  const int base = b * OUT_PER_B + iy * NXV + ix;
  // valid base < 2^23, so use sign bit as the keep flag: one readlane per point
  const int enc  = keep ? base : -1;

  const size_t pt0 = (size_t)bn * PTS_PER_BN + (size_t)grp * 16;
  const float* f0 = x_img + pt0 * CAMC;
  const int chOff = 2 * lane * CH_STRIDE;   // channels c = 2*lane, 2*lane+1

  #pragma unroll 4
  for (int pi = 0; pi < 16; ++pi) {
    const int e = __builtin_amdgcn_readlane(enc, pi);  // SGPR broadcast
    if (pi + 2 < 16) __builtin_prefetch(f0 + (pi + 2) * CAMC, 0, 1);
    if (e >= 0) {  // scalar branch, EXEC untouched
      const float2 fv = ((const float2*)(f0 + pi * CAMC))[lane];  // 256B/wave
      float* o = out + e + chOff;
      unsafeAtomicAdd(o, fv.x);
      unsafeAtomicAdd(o + CH_STRIDE, fv.y);
    }
  }
}

extern "C" void kernel_launch(void* const* d_in, const int* in_sizes, int n_in,
                              void* d_out, int out_size, void* d_ws, size_t ws_size,
                              hipStream_t stream) {
  const float* x_img      = (const float*)d_in[0];
  const float* rots       = (const float*)d_in[1];
  const float* trans      = (const float*)d_in[2];
  const float* intrins    = (const float*)d_in[3];
  const float* post_rots  = (const float*)d_in[4];
  const float* post_trans = (const float*)d_in[5];
  float* out = (float*)d_out;
  float* ws  = (float*)d_ws;

  lss_setup<<<1, 32, 0, stream>>>(rots, trans, intrins, post_rots, post_trans, ws);

  const int n4 = out_size / 4;  // 2,560,000 float4s
  lss_zero<<<(n4 + 255) / 256, 256, 0, stream>>>(out, n4);

  const int ngroups = NPRIME / 16;          // 43,296 waves
  const int blocks  = ngroups / 8;          // 5,412 blocks x 8 waves
  lss_scatter<<<blocks, 256, 0, stream>>>(x_img, ws, out);
}